// VGAE_83090437308757
// MI455X (gfx1250) — compile-verified
//
#include <hip/hip_runtime.h>
#include <math.h>

typedef __attribute__((ext_vector_type(2))) float v2f;
typedef __attribute__((ext_vector_type(8))) float v8f;

#define Nn 16384
#define Kk 16
#define Dd 256
#define EVv 128

// ======================================================================
// Generic fp32 WMMA GEMM:  C[M,Nout] = A[M,Kd] @ B  (+ biasScale*bias[col])
//                          (+ addSrc[row,col]);  Nout fixed via grid.y*64
// One wave computes a 16x64 strip of C (4 accumulator tiles, A reused).
// Uses V_WMMA_F32_16X16X4_F32 (fp32 in/out -> matches fp32 reference).
// ======================================================================
template <bool TRANSB, bool HAS_BIAS, bool HAS_ADD>
__global__ __launch_bounds__(128) void gemm_wmma_f32(
    const float* __restrict__ A, int lda,
    const float* __restrict__ B, int ldb,
    const float* __restrict__ bias, float biasScale,
    const float* __restrict__ addSrc,
    float* __restrict__ C, int ldc,
    int M, int Kd)
{
  const int lane = threadIdx.x & 31;
  const int wave = threadIdx.x >> 5;
  const int row0 = (blockIdx.x * 4 + wave) * 16;
  const int col0 = blockIdx.y * 64;
  if (row0 >= M) return;

  const int m  = lane & 15;          // A row within tile
  const int kq = (lane >> 4) << 1;   // A/B k-pair: 0 or 2
  const int nc = lane & 15;          // B/C column within tile

  v8f acc[4];
  acc[0] = {}; acc[1] = {}; acc[2] = {}; acc[3] = {};

  for (int kb = 0; kb < Kd; kb += 4) {
    // A fragment: a.x=A[m][kb+kq], a.y=A[m][kb+kq+1]
    v2f a = *(const v2f*)(A + (size_t)(row0 + m) * lda + kb + kq);
#pragma unroll
    for (int t = 0; t < 4; ++t) {
      const int c = col0 + t * 16 + nc;
      v2f b;
      if (TRANSB) {
        // B[k][n] = Braw[n][k]  (contiguous k,k+1 -> 8-byte load)
        b = *(const v2f*)(B + (size_t)c * ldb + kb + kq);
      } else {
        b.x = B[(size_t)(kb + kq)     * ldb + c];
        b.y = B[(size_t)(kb + kq + 1) * ldb + c];
      }
      acc[t] = __builtin_amdgcn_wmma_f32_16x16x4_f32(
          false, a, false, b, (short)0, acc[t], false, false);
    }
  }

  // C layout: VGPR r -> row (lane<16 ? r : r+8), col = lane%16
  const int rbase = (lane < 16) ? 0 : 8;
#pragma unroll
  for (int t = 0; t < 4; ++t) {
    const int c = col0 + t * 16 + nc;
#pragma unroll
    for (int r = 0; r < 8; ++r) {
      const int row = row0 + rbase + r;
      float v = acc[t][r];
      if (HAS_BIAS) v += biasScale * bias[c];
      if (HAS_ADD)  v += addSrc[(size_t)row * ldc + c];
      C[(size_t)row * ldc + c] = v;
    }
  }
}

// ======================================================================
// Gather + attention + neighbor aggregation. One wave32 per node.
// Each lane owns 8 contiguous dims of D=256.
// Writes R[n, 0:2D] = in_rep_sum + out_rep_sum  (both hit the same W_nb).
// ======================================================================
__global__ __launch_bounds__(256) void gather_attn(
    const float* __restrict__ x,      // [N,D]
    const float* __restrict__ Xn,     // [N,D]  x @ W_node_top
    const float* __restrict__ En,     // [EV,D] edge_emb @ W_node_bot + b_node
    const float* __restrict__ Y,      // [N,D]  x @ attenW^T
    const float* __restrict__ eemb,   // [EV,D]
    const int*   __restrict__ in_idx,  const int* __restrict__ in_edge,
    const float* __restrict__ in_mask,
    const int*   __restrict__ out_idx, const int* __restrict__ out_edge,
    const float* __restrict__ out_mask,
    float* __restrict__ R)            // [N, 2D]
{
  const int lane = threadIdx.x & 31;
  const int n = blockIdx.x * (blockDim.x >> 5) + (threadIdx.x >> 5);
  if (n >= Nn) return;
  const int seg = lane * 8;

  const float4 y0 = *(const float4*)(Y + (size_t)n * Dd + seg);
  const float4 y1 = *(const float4*)(Y + (size_t)n * Dd + seg + 4);

  int oi[Kk], oe[Kk];
  float lg[Kk];
#pragma unroll
  for (int k = 0; k < Kk; ++k) {
    oi[k] = out_idx [(size_t)n * Kk + k];
    oe[k] = out_edge[(size_t)n * Kk + k];
    const float* xr = Xn + (size_t)oi[k] * Dd + seg;
    const float* er = En + (size_t)oe[k] * Dd + seg;
    const float4 a0 = *(const float4*)xr,      a1 = *(const float4*)(xr + 4);
    const float4 b0 = *(const float4*)er,      b1 = *(const float4*)(er + 4);
    float p = (a0.x + b0.x) * y0.x + (a0.y + b0.y) * y0.y +
              (a0.z + b0.z) * y0.z + (a0.w + b0.w) * y0.w +
              (a1.x + b1.x) * y1.x + (a1.y + b1.y) * y1.y +
              (a1.z + b1.z) * y1.z + (a1.w + b1.w) * y1.w;
#pragma unroll
    for (int off = 16; off > 0; off >>= 1) p += __shfl_xor(p, off, 32);
    lg[k] = p;
  }

  // softmax over K=16 (redundant per-lane, all lanes identical)
  float mx = lg[0];
#pragma unroll
  for (int k = 1; k < Kk; ++k) mx = fmaxf(mx, lg[k]);
  float s = 0.f;
#pragma unroll
  for (int k = 0; k < Kk; ++k) { lg[k] = __expf(lg[k] - mx); s += lg[k]; }
  const float inv = 1.0f / s;

  float accN[8] = {0,0,0,0,0,0,0,0};
  float accE[8] = {0,0,0,0,0,0,0,0};

  // outgoing: alpha * mask weighted sum
#pragma unroll
  for (int k = 0; k < Kk; ++k) {
    const float w = lg[k] * inv * out_mask[(size_t)n * Kk + k];
    const float* xr = x    + (size_t)oi[k] * Dd + seg;
    const float* er = eemb + (size_t)oe[k] * Dd + seg;
    const float4 a0 = *(const float4*)xr, a1 = *(const float4*)(xr + 4);
    const float4 b0 = *(const float4*)er, b1 = *(const float4*)(er + 4);
    accN[0] = fmaf(w, a0.x, accN[0]); accN[1] = fmaf(w, a0.y, accN[1]);
    accN[2] = fmaf(w, a0.z, accN[2]); accN[3] = fmaf(w, a0.w, accN[3]);
    accN[4] = fmaf(w, a1.x, accN[4]); accN[5] = fmaf(w, a1.y, accN[5]);
    accN[6] = fmaf(w, a1.z, accN[6]); accN[7] = fmaf(w, a1.w, accN[7]);
    accE[0] = fmaf(w, b0.x, accE[0]); accE[1] = fmaf(w, b0.y, accE[1]);
    accE[2] = fmaf(w, b0.z, accE[2]); accE[3] = fmaf(w, b0.w, accE[3]);
    accE[4] = fmaf(w, b1.x, accE[4]); accE[5] = fmaf(w, b1.y, accE[5]);
    accE[6] = fmaf(w, b1.z, accE[6]); accE[7] = fmaf(w, b1.w, accE[7]);
  }

  // incoming: mask weighted sum
#pragma unroll
  for (int k = 0; k < Kk; ++k) {
    const float w = in_mask[(size_t)n * Kk + k];
    const int ii = in_idx [(size_t)n * Kk + k];
    const int ie = in_edge[(size_t)n * Kk + k];
    const float* xr = x    + (size_t)ii * Dd + seg;
    const float* er = eemb + (size_t)ie * Dd + seg;
    const float4 a0 = *(const float4*)xr, a1 = *(const float4*)(xr + 4);
    const float4 b0 = *(const float4*)er, b1 = *(const float4*)(er + 4);
    accN[0] = fmaf(w, a0.x, accN[0]); accN[1] = fmaf(w, a0.y, accN[1]);
    accN[2] = fmaf(w, a0.z, accN[2]); accN[3] = fmaf(w, a0.w, accN[3]);
    accN[4] = fmaf(w, a1.x, accN[4]); accN[5] = fmaf(w, a1.y, accN[5]);
    accN[6] = fmaf(w, a1.z, accN[6]); accN[7] = fmaf(w, a1.w, accN[7]);
    accE[0] = fmaf(w, b0.x, accE[0]); accE[1] = fmaf(w, b0.y, accE[1]);
    accE[2] = fmaf(w, b0.z, accE[2]); accE[3] = fmaf(w, b0.w, accE[3]);
    accE[4] = fmaf(w, b1.x, accE[4]); accE[5] = fmaf(w, b1.y, accE[5]);
    accE[6] = fmaf(w, b1.z, accE[6]); accE[7] = fmaf(w, b1.w, accE[7]);
  }

  float* rn = R + (size_t)n * (2 * Dd);
  *(float4*)(rn + seg)          = make_float4(accN[0], accN[1], accN[2], accN[3]);
  *(float4*)(rn + seg + 4)      = make_float4(accN[4], accN[5], accN[6], accN[7]);
  *(float4*)(rn + Dd + seg)     = make_float4(accE[0], accE[1], accE[2], accE[3]);
  *(float4*)(rn + Dd + seg + 4) = make_float4(accE[4], accE[5], accE[6], accE[7]);
}

// ======================================================================
// Readout stage 1: per-block partial sums of h[n,0:2] (pre-bias).
// h[n] = relu(NH[n] @ W1 + b1) @ W2.  W1 staged in LDS (64 KB of 320 KB).
// ======================================================================
__global__ __launch_bounds__(256) void readout_partial(
    const float* __restrict__ NH, const float* __restrict__ W1,
    const float* __restrict__ b1, const float* __restrict__ W2,
    float* __restrict__ partials)
{
  __shared__ float sW1[Dd * 64];
  __shared__ float s0[256], s1[256];
  for (int i = threadIdx.x; i < Dd * 64; i += 256) sW1[i] = W1[i];
  __syncthreads();

  const int n = blockIdx.x * 256 + threadIdx.x;
  float r[64];
#pragma unroll
  for (int j = 0; j < 64; ++j) r[j] = b1[j];
  const float* nh = NH + (size_t)n * Dd;
  for (int d = 0; d < Dd; ++d) {
    const float v = nh[d];
#pragma unroll
    for (int j = 0; j < 64; ++j) r[j] = fmaf(v, sW1[d * 64 + j], r[j]);
  }
  float p0 = 0.f, p1 = 0.f;
#pragma unroll
  for (int j = 0; j < 64; ++j) {
    const float rr = fmaxf(r[j], 0.f);
    p0 = fmaf(rr, W2[2 * j],     p0);
    p1 = fmaf(rr, W2[2 * j + 1], p1);
  }
  s0[threadIdx.x] = p0; s1[threadIdx.x] = p1;
  __syncthreads();
  for (int st = 128; st > 0; st >>= 1) {
    if (threadIdx.x < st) {
      s0[threadIdx.x] += s0[threadIdx.x + st];
      s1[threadIdx.x] += s1[threadIdx.x + st];
    }
    __syncthreads();
  }
  if (threadIdx.x == 0) {
    partials[blockIdx.x * 2]     = s0[0];
    partials[blockIdx.x * 2 + 1] = s1[0];
  }
}

// Readout stage 2: deterministic final reduce + 2-way softmax.
__global__ void readout_final(const float* __restrict__ partials, int nparts,
                              const float* __restrict__ b2,
                              float* __restrict__ out)
{
  if (threadIdx.x == 0 && blockIdx.x == 0) {
    float s0 = (float)Nn * b2[0];
    float s1 = (float)Nn * b2[1];
    for (int i = 0; i < nparts; ++i) { s0 += partials[2 * i]; s1 += partials[2 * i + 1]; }
    const float m  = fmaxf(s0, s1);
    const float e0 = __expf(s0 - m), e1 = __expf(s1 - m);
    const float is = 1.0f / (e0 + e1);
    out[0] = e0 * is;
    out[1] = e1 * is;
  }
}

// ======================================================================
extern "C" void kernel_launch(void* const* d_in, const int* in_sizes, int n_in,
                              void* d_out, int out_size, void* d_ws, size_t ws_size,
                              hipStream_t stream)
{
  const float* x        = (const float*)d_in[0];
  const int*   in_idx   = (const int*)  d_in[1];
  const int*   in_edge  = (const int*)  d_in[2];
  const float* in_mask  = (const float*)d_in[3];
  const int*   out_idx  = (const int*)  d_in[4];
  const int*   out_edge = (const int*)  d_in[5];
  const float* out_mask = (const float*)d_in[6];
  const float* eemb     = (const float*)d_in[7];
  const float* W_nb     = (const float*)d_in[8];
  const float* b_nb     = (const float*)d_in[9];
  const float* W_node   = (const float*)d_in[10];
  const float* b_node   = (const float*)d_in[11];
  const float* attenW   = (const float*)d_in[12];
  const float* W1       = (const float*)d_in[13];
  const float* b1       = (const float*)d_in[14];
  const float* W2       = (const float*)d_in[15];
  const float* b2       = (const float*)d_in[16];
  (void)in_sizes; (void)n_in; (void)out_size; (void)ws_size;

  // workspace layout (floats): En[128*256] Xn[N*256] Y[N*256] R[N*512] partials[128]
  float* ws = (float*)d_ws;
  float* En = ws;
  float* Xn = En + (size_t)EVv * Dd;
  float* Y  = Xn + (size_t)Nn * Dd;
  float* R  = Y  + (size_t)Nn * Dd;
  float* partials = R + (size_t)Nn * 2 * Dd;

  float* NH      = (float*)d_out;          // [N,D] node_hidden
  float* readout = NH + (size_t)Nn * Dd;   // [2]

  const dim3 blk(128);
  const dim3 gridBig(Nn / 64, Dd / 64);

  // En' = edge_emb @ W_node[D:,:] + b_node   (128 x 256 x 256)
  gemm_wmma_f32<false, true, false><<<dim3(EVv / 64, Dd / 64), blk, 0, stream>>>(
      eemb, Dd, W_node + (size_t)Dd * Dd, Dd, b_node, 1.0f, nullptr, En, Dd, EVv, Dd);
  // Xn = x @ W_node[:D,:]                    (16384 x 256 x 256)
  gemm_wmma_f32<false, false, false><<<gridBig, blk, 0, stream>>>(
      x, Dd, W_node, Dd, nullptr, 0.f, nullptr, Xn, Dd, Nn, Dd);
  // Y = x @ attenW^T                         (16384 x 256 x 256)
  gemm_wmma_f32<true, false, false><<<gridBig, blk, 0, stream>>>(
      x, Dd, attenW, Dd, nullptr, 0.f, nullptr, Y, Dd, Nn, Dd);

  // gather + attention + aggregation: one wave per node
  gather_attn<<<Nn / 8, 256, 0, stream>>>(x, Xn, En, Y, eemb,
                                          in_idx, in_edge, in_mask,
                                          out_idx, out_edge, out_mask, R);

  // node_hidden = x + R @ W_nb + 2*b_nb      (16384 x 512 x 256)
  gemm_wmma_f32<false, true, true><<<gridBig, blk, 0, stream>>>(
      R, 2 * Dd, W_nb, Dd, b_nb, 2.0f, x, NH, Dd, Nn, 2 * Dd);

  // readout
  readout_partial<<<Nn / 256, 256, 0, stream>>>(NH, W1, b1, W2, partials);
  readout_final<<<1, 32, 0, stream>>>(partials, Nn / 256, b2, readout);
}